// LGNCompact_25288767439404
// MI455X (gfx1250) — compile-verified
//
#include <hip/hip_runtime.h>

// ---------------------------------------------------------------------------
// LGN compact: prefix-product of 2x2 matrix exponentials of a Magnus step.
//   T = 524289 samples, NI = 524288 intervals, out = (T,2) f32.
// Pipeline (all f32):
//   K1: per-wave segment totals (E recomputed in regs, never spilled to HBM)
//   K2: 1-block exclusive scan of 4096 wave totals -> carries (128 KB ws)
//   K3: re-walk segments, emit x_{i+1} = (E_i ... E_0) x0
// A(t) evaluation uses V_WMMA_F32_16X16X4_F32: out[4 x 16pts] = W[4x52] @ F[52x16pts]
// as 13 accumulated K=4 f32 WMMAs (full fp32 -> numerically faithful).
// Trig: custom Cody-Waite pi/2 reduction (phases provably <= 252 rad), so no
// ocml Payne-Hanek dead code. expm uses entire-function series in s2.
// ---------------------------------------------------------------------------

typedef float v2f __attribute__((ext_vector_type(2)));
typedef float v8f __attribute__((ext_vector_type(8)));

#define NI      524288       // number of intervals (T-1)
#define TLAST   524288       // last valid index into t[]
#define ITERS   8            // iterations per wave, 16 intervals each
#define NBLK    512
#define WPB     8            // waves per 256-thread block
#define NWAVES  (NBLK * WPB) // 4096 ; NWAVES * 16 * ITERS == NI

__device__ __forceinline__ float4 mk4(float a, float b, float c, float d) {
    float4 r; r.x = a; r.y = b; r.z = c; r.w = d; return r;
}

// 2x2 matmul, row-major packed as {m00,m01,m10,m11}
__device__ __forceinline__ float4 mmul(const float4 X, const float4 Y) {
    return mk4(X.x * Y.x + X.y * Y.z, X.x * Y.y + X.y * Y.w,
               X.z * Y.x + X.w * Y.z, X.z * Y.y + X.w * Y.w);
}

__device__ __forceinline__ float4 shfl4(float4 v, int l) {
    return mk4(__shfl(v.x, l, 32), __shfl(v.y, l, 32),
               __shfl(v.z, l, 32), __shfl(v.w, l, 32));
}
__device__ __forceinline__ float4 shflup4(float4 v, unsigned d) {
    return mk4(__shfl_up(v.x, d, 32), __shfl_up(v.y, d, 32),
               __shfl_up(v.z, d, 32), __shfl_up(v.w, d, 32));
}
__device__ __forceinline__ float4 shfldn4(float4 v, unsigned d) {
    return mk4(__shfl_down(v.x, d, 32), __shfl_down(v.y, d, 32),
               __shfl_down(v.z, d, 32), __shfl_down(v.w, d, 32));
}

// sin/cos for |x| <= ~400 rad: 4-term Cody-Waite pi/2 reduction (q <= 255,
// q*PI2_A exact: 8-bit q x 7-bit mantissa), minimax polys on [-pi/4, pi/4],
// branchless quadrant select. ~1 ulp; avoids ocml Payne-Hanek code bloat.
__device__ __forceinline__ float trigf(float x, bool wantCos) {
    float q = rintf(x * 0.636619772367581343f);   // x * 2/pi
    int qi = (int)q;
    float r = fmaf(q, -1.5703125f, x);
    r = fmaf(q, -4.8351287841796875e-4f, r);
    r = fmaf(q, -3.13855707645416259766e-7f, r);
    r = fmaf(q, -6.07710062827671038e-11f, r);
    float s = r * r;
    // sin(r) = r * (1 + s*(S1 + s*(S2 + s*(S3 + s*S4))))
    float sp = fmaf(s, 2.7557314297e-06f, -1.9841270114e-04f);
    sp = fmaf(s, sp, 8.3333337680e-03f);
    sp = fmaf(s, sp, -1.6666667163e-01f);
    sp = r * fmaf(s, sp, 1.0f);
    // cos(r) = 1 + s*(C1 + s*(C2 + s*(C3 + s*C4)))
    float cp = fmaf(s, 2.4801587642e-05f, -1.3888889225e-03f);
    cp = fmaf(s, cp, 4.1666667908e-02f);
    cp = fmaf(s, cp, -0.5f);
    cp = fmaf(s, cp, 1.0f);
    int qq = qi + (wantCos ? 1 : 0);
    float v = (qq & 1) ? cp : sp;
    return (qq & 2) ? -v : v;
}

// Fourier feature row r at time tp: rows 0..24 = cos(tp*(r+1)),
// rows 25..49 = sin(tp*(r-24)), rows 50..51 = 0 (K padding to 52).
__device__ __forceinline__ float rowval(int r, float tp) {
    bool isCos = (r < 25);
    float freq = isCos ? (float)(r + 1) : (float)(r - 24);
    float v = trigf(tp * freq, isCos);
    return (r < 50) ? v : 0.0f;
}

// One 16-point batch of A(t) via 13x V_WMMA_F32_16X16X4_F32.
// Lane (h = lane>>4, p = lane&15) supplies, per K-chunk c:
//   A tile (W):   v0 = W[p][4c+2h],   v1 = W[p][4c+2h+1]   (p<4 else 0)
//   B tile (F):   v0 = F[4c+2h][p],   v1 = F[4c+2h+1][p]
// Result rows M=0..3 land in acc[0..3] on lanes 0..15 (N = p).
__device__ __forceinline__ float4 evalA(float tp, const v2f* wA,
                                        const float4 bias, int h) {
    v8f acc = {0.f, 0.f, 0.f, 0.f, 0.f, 0.f, 0.f, 0.f};
#pragma unroll
    for (int c = 0; c < 13; ++c) {
        int r0 = 4 * c + 2 * h;
        v2f B;
        B.x = rowval(r0, tp);
        B.y = rowval(r0 + 1, tp);
        acc = __builtin_amdgcn_wmma_f32_16x16x4_f32(
            false, wA[c], false, B, (short)0, acc, false, false);
    }
    return mk4(acc[0] + bias.x, acc[1] + bias.y,
               acc[2] + bias.z, acc[3] + bias.w);
}

// Closed-form exp of 2x2. cosh(sqrt(s2)) and sinh(sqrt(s2))/sqrt(s2) are
// entire functions of s2 (series in s2 covers hyperbolic s2>0, rotational
// s2<0, and the tiny-s2 Taylor branch of the reference uniformly).
// Here |s2| ~ (|A| dt)^2 ~ 1e-9, series is exact to f32; valid to |s2|<~1.
__device__ __forceinline__ float4 expm2x2(const float4 M) {
    float mu = 0.5f * (M.x + M.w);
    float pp = 0.5f * (M.x - M.w);
    float s2 = pp * pp + M.y * M.z;
    float ch = fmaf(s2, 2.48015873015873016e-5f, 1.38888888888888889e-3f);
    ch = fmaf(s2, ch, 4.16666666666666667e-2f);
    ch = fmaf(s2, ch, 0.5f);
    ch = fmaf(s2, ch, 1.0f);
    float sh = fmaf(s2, 2.75573192239858907e-6f, 1.98412698412698413e-4f);
    sh = fmaf(s2, sh, 8.33333333333333333e-3f);
    sh = fmaf(s2, sh, 1.66666666666666667e-1f);
    sh = fmaf(s2, sh, 1.0f);
    float em = expf(mu);
    return mk4(em * (ch + sh * pp), em * (sh * M.y),
               em * (sh * M.z),     em * (ch - sh * pp));
}

// Shared wave pass. EMIT=false: accumulate segment total -> waveIO[w].
// EMIT=true: waveIO[w] is the exclusive carry; emit outputs.
template <bool EMIT>
__device__ __forceinline__ void wave_run(const float* __restrict__ t,
                                         const float* __restrict__ Wm,
                                         const float* __restrict__ bvec,
                                         const float* __restrict__ x0,
                                         float4* __restrict__ waveIO,
                                         float* __restrict__ out) {
    const int tid  = threadIdx.x;
    const int lane = tid & 31;
    const int h    = lane >> 4;
    const int p    = lane & 15;
    const int w    = blockIdx.x * WPB + (tid >> 5);
    const int s0   = w * (16 * ITERS);

    // Preload W tiles for all 13 K-chunks (A operand of each WMMA).
    v2f wA[13];
#pragma unroll
    for (int c = 0; c < 13; ++c) {
        int r0 = 4 * c + 2 * h;
        float w0 = (p < 4 && r0 < 50)       ? Wm[p * 50 + r0]     : 0.0f;
        float w1 = (p < 4 && (r0 + 1) < 50) ? Wm[p * 50 + r0 + 1] : 0.0f;
        v2f wv; wv.x = w0; wv.y = w1; wA[c] = wv;
    }
    const float4 bias = mk4(bvec[0], bvec[1], bvec[2], bvec[3]);
    const float4 I    = mk4(1.f, 0.f, 0.f, 1.f);

    float4 Wtot = I;
    float2 v; v.x = 0.f; v.y = 0.f;
    if (EMIT) {
        float4 C = waveIO[w];
        float a0 = x0[0], a1 = x0[1];
        v.x = C.x * a0 + C.y * a1;
        v.y = C.z * a0 + C.w * a1;
    }

    // Pipelined endpoint A's: Aend holds A(t[base+p]).
    float4 Aend = evalA(t[s0 + p], wA, bias, h);

#pragma unroll 1
    for (int it = 0; it < ITERS; ++it) {
        const int base = s0 + 16 * it;
        const float t0p = t[base + p];       // <= 524287
        const float t1p = t[base + p + 1];   // <= 524288
        const float tm  = 0.5f * (t0p + t1p);

        float4 Amid = evalA(tm, wA, bias, h);
        int nidx = base + 16 + p;
        if (nidx > TLAST) nidx = TLAST;      // tail clamp (values unused)
        float4 Anext = evalA(t[nidx], wA, bias, h);

        // A1 for lane p is Aend of lane p+1; lane 15 takes Anext lane 0.
        float4 A1s    = shfldn4(Aend, 1);
        float4 Afirst = shfl4(Anext, 0);
        float4 A1 = (p == 15) ? Afirst : A1s;

        float dt = t1p - t0p;
        float4 P1 = mmul(Aend, A1);
        float4 P2 = mmul(A1, Aend);
        float kk = dt * dt / 12.0f;
        float4 Om = mk4(Amid.x * dt + kk * (P1.x - P2.x),
                        Amid.y * dt + kk * (P1.y - P2.y),
                        Amid.z * dt + kk * (P1.z - P2.z),
                        Amid.w * dt + kk * (P1.w - P2.w));
        float4 E = expm2x2(Om);
        if (h == 1) E = I;   // identity-pad upper half-wave

        // Ordered inclusive scan over lanes: S_p = E_p @ ... @ E_0.
        float4 S = E;
#pragma unroll
        for (int off = 1; off < 32; off <<= 1) {
            float4 o = shflup4(S, off);
            if (lane >= (unsigned)off) S = mmul(S, o);
        }

        if (EMIT && h == 0) {
            int oi = (base + p + 1) * 2;
            out[oi]     = S.x * v.x + S.y * v.y;
            out[oi + 1] = S.z * v.x + S.w * v.y;
        }
        float4 Tt = shfl4(S, 15);   // full 16-interval product
        if (EMIT) {
            float nx = Tt.x * v.x + Tt.y * v.y;
            float ny = Tt.z * v.x + Tt.w * v.y;
            v.x = nx; v.y = ny;
        } else {
            Wtot = mmul(Tt, Wtot);
        }
        Aend = Anext;
    }

    if (!EMIT && lane == 0) waveIO[w] = Wtot;
}

__global__ void __launch_bounds__(256)
k_wave_totals(const float* __restrict__ t, const float* __restrict__ Wm,
              const float* __restrict__ b, float4* __restrict__ waveTot) {
    wave_run<false>(t, Wm, b, nullptr, waveTot, nullptr);
}

__global__ void __launch_bounds__(256)
k_scan_carries(const float4* __restrict__ tot, float4* __restrict__ carry) {
    __shared__ float4 lds[256];
    const int ti = threadIdx.x;
    const float4 I = mk4(1.f, 0.f, 0.f, 1.f);
    float4 P = I;
    for (int j = 0; j < 16; ++j) P = mmul(tot[ti * 16 + j], P);
    lds[ti] = P;
    __syncthreads();
    float4 incl = P;
    for (int off = 1; off < 256; off <<= 1) {
        float4 o = (ti >= off) ? lds[ti - off] : I;
        __syncthreads();
        if (ti >= off) { incl = mmul(incl, o); lds[ti] = incl; }
        __syncthreads();
    }
    float4 run = (ti == 0) ? I : lds[ti - 1];   // exclusive prefix
    for (int j = 0; j < 16; ++j) {
        carry[ti * 16 + j] = run;
        run = mmul(tot[ti * 16 + j], run);
    }
}

__global__ void __launch_bounds__(256)
k_emit(const float* __restrict__ t, const float* __restrict__ Wm,
       const float* __restrict__ b, const float* __restrict__ x0,
       float4* __restrict__ waveCarry, float* __restrict__ out) {
    wave_run<true>(t, Wm, b, x0, waveCarry, out);
    if (blockIdx.x == 0 && threadIdx.x == 0) {
        out[0] = x0[0];
        out[1] = x0[1];
    }
}

extern "C" void kernel_launch(void* const* d_in, const int* in_sizes, int n_in,
                              void* d_out, int out_size, void* d_ws,
                              size_t ws_size, hipStream_t stream) {
    (void)in_sizes; (void)n_in; (void)out_size; (void)ws_size;
    const float* t  = (const float*)d_in[0];
    const float* x0 = (const float*)d_in[1];
    const float* Wm = (const float*)d_in[3];
    const float* b  = (const float*)d_in[4];
    float* out = (float*)d_out;

    float4* waveTot   = (float4*)d_ws;          // 4096 * 16 B
    float4* waveCarry = waveTot + NWAVES;       // 4096 * 16 B (128 KB total)

    k_wave_totals<<<NBLK, 256, 0, stream>>>(t, Wm, b, waveTot);
    k_scan_carries<<<1, 256, 0, stream>>>(waveTot, waveCarry);
    k_emit<<<NBLK, 256, 0, stream>>>(t, Wm, b, x0, waveCarry, out);
}